// HamEvo_52767968198729
// MI455X (gfx1250) — compile-verified
//
#include <hip/hip_runtime.h>

// HamEvo RK4 on MI455X (gfx1250): one workgroup per batch element.
// state slice (64x128 complex) + H_b (64x64 complex) stay on-chip (LDS+VGPR)
// for all 100 RK4 steps; complex matmul via V_WMMA_F32_16X16X4_F32.

typedef __attribute__((ext_vector_type(2))) float v2f;
typedef __attribute__((ext_vector_type(8))) float v8f;

#define DIM     64     // 2^6 target-qubit dimension (contracted)
#define RCOLS   128    // 2^7 spectator dimension
#define NBATCH  128
#define NSTEPS  100

// LDS layout (floats), planes of 8192 floats (64x128):
//   [ub + 0]      U_re     ub in {0, 16384} (ping-pong gate input)
//   [ub + 8192]   U_im
//   [32768]       S_re     persistent state
//   [40960]       S_im
//   [49152]       ACC_re   RK4 accumulator (k1 + 2k2 + 2k3)
//   [57344]       ACC_im
#define UBUF_STRIDE 16384
#define IM_OFF      8192
#define S_OFF       32768
#define ACC_OFF     49152
#define SMEM_FLOATS 65536   // 256 KB

static __device__ __forceinline__ v8f wmma_f32(v2f a, v2f b, v8f c) {
  // D = A(16x4 f32) * B(4x16 f32) + C(16x16 f32)
  return __builtin_amdgcn_wmma_f32_16x16x4_f32(false, a, false, b,
                                               (short)0, c, false, false);
}

__global__ void __launch_bounds__(256)
hamevo_rk4_kernel(const float* __restrict__ gS,   // state: [64][128][128] cplx interleaved
                  const float* __restrict__ gH,   // H:     [64][64][128]  cplx interleaved
                  const float* __restrict__ gT,   // t:     [128] f32
                  float* __restrict__ gOut)       // same layout as gS
{
  extern __shared__ float smem[];

  const int b    = blockIdx.x;
  const int tid  = threadIdx.x;
  const int wave = tid >> 5;       // 0..7 (wave32)
  const int lane = tid & 31;
  const int half = lane >> 4;      // 0: lanes 0-15, 1: lanes 16-31
  const int lan  = lane & 15;

  const int m0    = (wave >> 1) << 4;   // M-tile row base: 0,16,32,48
  const int ncol0 = (wave & 1) << 6;    // N column base: 0 or 64 (4 tiles of 16)

  const float h  = gT[b] * (1.0f / (float)NSTEPS);
  const float h2 = 0.5f * h;
  const float h6 = h * (1.0f / 6.0f);

  // ---- A = -i * H_b fragments, resident in VGPRs for the whole kernel ----
  // A_re = H_im ; A_im = -H_re ; nAim = +H_re (pre-negated so both complex
  // output parts are pure WMMA accumulate chains).
  // A 16x4 layout: lane -> M = m0+lan ; VGPR j -> K = 4*kk + 2*half + j.
  v2f aRe[16], aIm[16], aNim[16];
  {
    const int m = m0 + lan;
#pragma unroll
    for (int kk = 0; kk < 16; ++kk) {
      const int k  = kk * 4 + 2 * half;
      const int o0 = 2 * ((m * DIM + k) * NBATCH + b);  // (m,k)
      const int o1 = o0 + 2 * NBATCH;                   // (m,k+1)
      const float hre0 = gH[o0], him0 = gH[o0 + 1];
      const float hre1 = gH[o1], him1 = gH[o1 + 1];
      aRe[kk].x  =  him0; aRe[kk].y  =  him1;
      aIm[kk].x  = -hre0; aIm[kk].y  = -hre1;
      aNim[kk].x =  hre0; aNim[kk].y =  hre1;
    }
  }

  // ---- Load this wave's state tiles into LDS: S plane and U0 = s ----
  // C/D 16x16 layout: VGPR r -> M = m0 + r + 8*half ; lane -> N = n.
  // Each (m,n) element belongs to exactly one wave -> no cross-wave S/ACC access.
#pragma unroll
  for (int nt = 0; nt < 4; ++nt) {
    const int n = ncol0 + nt * 16 + lan;
#pragma unroll
    for (int r = 0; r < 8; ++r) {
      const int m   = m0 + r + 8 * half;
      const int o   = 2 * ((m * RCOLS + n) * NBATCH + b);
      const int idx = m * RCOLS + n;
      const float vr = gS[o], vi = gS[o + 1];
      smem[S_OFF + idx]          = vr;
      smem[S_OFF + IM_OFF + idx] = vi;
      smem[idx]                  = vr;   // U0_re
      smem[IM_OFF + idx]         = vi;   // U0_im
    }
  }
  __syncthreads();

#pragma unroll 1
  for (int step = 0; step < NSTEPS; ++step) {
    int ub = 0;                            // float offset of current U buffer
#pragma unroll 1
    for (int g = 0; g < 4; ++g) {          // k1..k4 gate evaluations
      const int vb = ub ^ UBUF_STRIDE;     // other (write) buffer
      const float c = (g < 2) ? h2 : h;    // coeff building next gate input
      // unroll 2 -> 4 independent WMMA accumulate chains on the matrix pipe
#pragma unroll 2
      for (int nt = 0; nt < 4; ++nt) {
        const int n = ncol0 + nt * 16 + lan;
        v8f kRe = {0.f,0.f,0.f,0.f,0.f,0.f,0.f,0.f};
        v8f kIm = {0.f,0.f,0.f,0.f,0.f,0.f,0.f,0.f};
        // K = gate(U): complex 16x64 @ 64x16 via 4 real WMMA chains
#pragma unroll
        for (int kk = 0; kk < 16; ++kk) {
          const int krow = kk * 4 + 2 * half;  // B 4x16: lane->N, half->K pair
          const int o0 = ub + krow * RCOLS + n;
          v2f bRe, bIm;
          bRe.x = smem[o0];
          bRe.y = smem[o0 + RCOLS];
          bIm.x = smem[o0 + IM_OFF];
          bIm.y = smem[o0 + IM_OFF + RCOLS];
          kRe = wmma_f32(aRe[kk],  bRe, kRe);   // + Him*Ure
          kRe = wmma_f32(aNim[kk], bIm, kRe);   // + Hre*Uim
          kIm = wmma_f32(aRe[kk],  bIm, kIm);   // + Him*Uim
          kIm = wmma_f32(aIm[kk],  bRe, kIm);   // - Hre*Ure
        }
        // Gate tails (ACC/S tiles are wave-private; addresses are LDS so a
        // runtime nt costs only address math, never register indexing):
        if (g == 0) {
          // ACC = k1 ; U' = S + h2*k1
#pragma unroll
          for (int r = 0; r < 8; ++r) {
            const int m   = m0 + r + 8 * half;
            const int idx = m * RCOLS + n;
            smem[ACC_OFF + idx]          = kRe[r];
            smem[ACC_OFF + IM_OFF + idx] = kIm[r];
            smem[vb + idx]          = smem[S_OFF + idx]          + c * kRe[r];
            smem[vb + IM_OFF + idx] = smem[S_OFF + IM_OFF + idx] + c * kIm[r];
          }
        } else if (g < 3) {
          // ACC += 2*k{2,3} ; U' = S + c*k
#pragma unroll
          for (int r = 0; r < 8; ++r) {
            const int m   = m0 + r + 8 * half;
            const int idx = m * RCOLS + n;
            smem[ACC_OFF + idx]          += 2.0f * kRe[r];
            smem[ACC_OFF + IM_OFF + idx] += 2.0f * kIm[r];
            smem[vb + idx]          = smem[S_OFF + idx]          + c * kRe[r];
            smem[vb + IM_OFF + idx] = smem[S_OFF + IM_OFF + idx] + c * kIm[r];
          }
        } else {
          // g==3: S_new = S + (h/6)*(ACC + k4); write S and U0 (vb==U0 here)
#pragma unroll
          for (int r = 0; r < 8; ++r) {
            const int m   = m0 + r + 8 * half;
            const int idx = m * RCOLS + n;
            const float vr = smem[S_OFF + idx]
                           + h6 * (smem[ACC_OFF + idx] + kRe[r]);
            const float vi = smem[S_OFF + IM_OFF + idx]
                           + h6 * (smem[ACC_OFF + IM_OFF + idx] + kIm[r]);
            smem[S_OFF + idx]          = vr;
            smem[S_OFF + IM_OFF + idx] = vi;
            smem[vb + idx]             = vr;   // U0_re for next step
            smem[vb + IM_OFF + idx]    = vi;   // U0_im
          }
        }
      }
      __syncthreads();   // ping-pong buffers -> one barrier per gate
      ub ^= UBUF_STRIDE;
    }
  }

  // ---- Store final state ----
#pragma unroll
  for (int nt = 0; nt < 4; ++nt) {
    const int n = ncol0 + nt * 16 + lan;
#pragma unroll
    for (int r = 0; r < 8; ++r) {
      const int m   = m0 + r + 8 * half;
      const int idx = m * RCOLS + n;
      const int o   = 2 * ((m * RCOLS + n) * NBATCH + b);
      gOut[o]     = smem[S_OFF + idx];
      gOut[o + 1] = smem[S_OFF + IM_OFF + idx];
    }
  }
}

extern "C" void kernel_launch(void* const* d_in, const int* in_sizes, int n_in,
                              void* d_out, int out_size, void* d_ws, size_t ws_size,
                              hipStream_t stream) {
  (void)in_sizes; (void)n_in; (void)out_size; (void)d_ws; (void)ws_size;
  const float* gS = (const float*)d_in[0];  // complex64 state, interleaved floats
  const float* gH = (const float*)d_in[1];  // complex64 H, interleaved floats
  const float* gT = (const float*)d_in[2];  // float32 t
  float* gOut = (float*)d_out;

  const size_t smem_bytes = SMEM_FLOATS * sizeof(float);  // 256 KB dynamic LDS
  hipError_t err = hipFuncSetAttribute((const void*)hamevo_rk4_kernel,
                                       hipFuncAttributeMaxDynamicSharedMemorySize,
                                       (int)smem_bytes);
  (void)err;
  hamevo_rk4_kernel<<<dim3(NBATCH), dim3(256), smem_bytes, stream>>>(gS, gH, gT, gOut);
}